// QuantumGate_65481071400009
// MI455X (gfx1250) — compile-verified
//
#include <hip/hip_runtime.h>

// Quantum circuit simulator, 16 qubits, one batch element per workgroup.
// State part (real or imag, 64K floats = 256 KB) lives entirely in CDNA5 LDS.
// RY kron-fused 4-qubit unitaries applied via V_WMMA_F32_16X16X4_F32.
// CNOT ring applied as in-place vectorized (b128) LDS involution swaps.

#define N_QUBITS 16
#define N_LAYERS 2
#define BATCH    512
#define DIM      65536
#define THREADS  512   // 16 wave32 waves per workgroup

typedef __attribute__((ext_vector_type(2))) float v2f;
typedef __attribute__((ext_vector_type(8))) float v8f;

__global__ __launch_bounds__(THREADS)
void qsim16_kernel(const float* __restrict__ x,
                   const float* __restrict__ params,
                   float* __restrict__ out)
{
    // ---- LDS: 65536 state + fused-unitary tables + prefix tables ----
    __shared__ __align__(16) float S[DIM];         // 256 KB: one part of the state
    __shared__ float U[N_LAYERS][4][16][16];       // 8 KB: kron-fused 4-qubit RY unitaries
    __shared__ float Phi[256];                     // product of wires 0..7 factors (row r)
    __shared__ float Plo[256];                     // product of wires 8..15 factors (col c)
    __shared__ float cw[16], sw[16];               // RX half-angle cos/sin per wire
    __shared__ float pcs[N_LAYERS][16];            // RY half-angle cos per layer/wire
    __shared__ float pss[N_LAYERS][16];            // RY half-angle sin per layer/wire
    __shared__ float partial[16][16];              // per-wave expval partials

    float4* S4 = reinterpret_cast<float4*>(S);

    const int tid  = threadIdx.x;
    const int lane = tid & 31;
    const int wave = tid >> 5;                     // 0..15
    const int b    = blockIdx.x;

    // ================= Phase 0: per-batch tables =================
    if (tid < 16) {
        float a = 0.5f * x[b * N_QUBITS + tid];
        float s, c;
        __sincosf(a, &s, &c);
        cw[tid] = c;
        sw[tid] = s;
    } else if (tid < 48) {
        int idx = tid - 16;                        // 0..31
        int l = idx >> 4, w = idx & 15;
        float a = 0.5f * params[idx];
        float s, c;
        __sincosf(a, &s, &c);
        pcs[l][w] = c;
        pss[l][w] = s;
    }
    __syncthreads();

    // Prefix product tables for the RX product state.
    // wire w <-> z bit (15-w). Row r = z>>8 (wires 0..7), col c = z&255 (wires 8..15).
    if (tid < 256) {
        int r = tid;
        float p = 1.0f;
#pragma unroll
        for (int w = 0; w < 8; ++w)
            p *= ((r >> (7 - w)) & 1) ? sw[w] : cw[w];
        Phi[r] = p;
    } else {
        int c = tid - 256;
        float p = 1.0f;
#pragma unroll
        for (int w = 8; w < 16; ++w)
            p *= ((c >> (15 - w)) & 1) ? sw[w] : cw[w];
        Plo[c] = p;
    }

    // Fused 4-qubit RY unitaries: U[l][g] = RY(th_{l,4g}) (x) ... (x) RY(th_{l,4g+3})
    for (int e = tid; e < N_LAYERS * 4 * 256; e += THREADS) {
        int l  = e >> 10;
        int g  = (e >> 8) & 3;
        int ar = (e >> 4) & 15;                    // output axis index a'
        int ac = e & 15;                           // input axis index a
        float v = 1.0f;
#pragma unroll
        for (int m = 0; m < 4; ++m) {
            int w = 4 * g + m;
            int i = (ar >> (3 - m)) & 1;
            int j = (ac >> (3 - m)) & 1;
            float f = (i == j) ? pcs[l][w] : (i ? pss[l][w] : -pss[l][w]);
            v *= f;
        }
        U[l][g][ar][ac] = v;
    }
    __syncthreads();

    // ================= Evolve Re part then Im part =================
    float acc[16];
#pragma unroll
    for (int w = 0; w < 16; ++w) acc[w] = 0.0f;

    const int M  = lane & 15;                      // row (A) / col (B,C,D) index
    const int hs = lane >> 4;                      // lane half select

    for (int part = 0; part < 2; ++part) {
        // ---- Fill: product state amplitude part. amp(z) = m(z) * (-i)^popc(z)
        for (int jj = 0; jj < DIM / 4 / THREADS; ++jj) {   // 32 float4 blocks/thread
            int blkz = tid + THREADS * jj;         // consecutive lanes: conflict-free
            int z0   = blkz << 2;
            float ph_hi = Phi[z0 >> 8];
            int   k0   = __popc(z0);               // low 2 bits of z0 are zero
            float4 v;
#pragma unroll
            for (int e = 0; e < 4; ++e) {
                int k = (k0 + (e == 3 ? 2 : (e == 0 ? 0 : 1))) & 3;
                float ph;
                if (part == 0) ph = (k == 0) ? 1.0f : ((k == 2) ? -1.0f : 0.0f);
                else           ph = (k == 3) ? 1.0f : ((k == 1) ? -1.0f : 0.0f);
                float m = ph_hi * Plo[(z0 & 255) | e] * ph;
                if (e == 0) v.x = m; else if (e == 1) v.y = m;
                else if (e == 2) v.z = m; else v.w = m;
            }
            S4[blkz] = v;
        }
        __syncthreads();

        for (int l = 0; l < N_LAYERS; ++l) {
            // ---- RY layer: four 4-qubit group GEMMs via f32 WMMA, in-place in LDS
            for (int g = 0; g < 4; ++g) {
                const int beta = 12 - 4 * g;       // axis bit offset in z
                const float* Um = &U[l][g][0][0];

                // A fragment (16x4 tiles of U): lane M = lane&15, K = 4*k0 + 2*hs + v
                v2f afr[4];
#pragma unroll
                for (int k0 = 0; k0 < 4; ++k0) {
                    int K = 4 * k0 + 2 * hs;
                    afr[k0].x = Um[M * 16 + K];
                    afr[k0].y = Um[M * 16 + K + 1];
                }

                const int lowmask = (1 << beta) - 1;
                // 256 disjoint 16-fiber blocks; in-place safe, no barrier inside
                for (int blk = wave; blk < 256; blk += 16) {
                    int j  = blk * 16 + M;                          // fiber (12-bit)
                    int zb = ((j & ~lowmask) << 4) | (j & lowmask); // z with axis a=0
                    v8f c8 = {};
#pragma unroll
                    for (int k0 = 0; k0 < 4; ++k0) {
                        int K = 4 * k0 + 2 * hs;
                        v2f bfr;
                        bfr.x = S[zb | (K << beta)];
                        bfr.y = S[zb | ((K + 1) << beta)];
                        c8 = __builtin_amdgcn_wmma_f32_16x16x4_f32(
                                 false, afr[k0], false, bfr,
                                 (short)0, c8, false, false);
                    }
#pragma unroll
                    for (int v = 0; v < 8; ++v) {
                        int Mr = v + 8 * hs;                        // output axis a'
                        S[zb | (Mr << beta)] = c8[v];
                    }
                }
                __syncthreads();
            }

            // ---- CNOT ring: 16 in-place involution swap passes (vectorized b128)
            // Gates i=0..14: ctrl wire i -> tgt wire i+1  (bits 15-i -> 14-i)
            // Gate  i=15  : ctrl wire 15 -> tgt wire 0    (bit 0 -> bit 15)
            for (int i = 0; i < 13; ++i) {
                // ctrl bit = plo+1, tgt bit = plo, plo >= 2: whole float4 blocks swap
                const int plo = 14 - i;
                for (int qb = tid; qb < 4096; qb += THREADS) {
                    int q    = qb << 2;
                    int low  = q & ((1 << plo) - 1);
                    int high = q >> plo;
                    int z    = (high << (plo + 2)) | low | (1 << (plo + 1));
                    int i1   = z >> 2;
                    int i2   = i1 ^ (1 << (plo - 2));
                    float4 a = S4[i1];
                    float4 c = S4[i2];
                    S4[i1] = c;
                    S4[i2] = a;
                }
                __syncthreads();
            }
            {   // i=13: ctrl bit2, tgt bit1: blocks with bit2=1, swap (x,y)<->(z,w)
                for (int h = tid; h < 8192; h += THREADS) {
                    int idx  = ((h << 1) | 1);     // float4 index: z = idx*4, bit2=1
                    float4 v = S4[idx];
                    S4[idx]  = make_float4(v.z, v.w, v.x, v.y);
                }
                __syncthreads();
            }
            {   // i=14: ctrl bit1, tgt bit0: in-block swap of elements 2,3
                for (int idx = tid; idx < 16384; idx += THREADS) {
                    float4 v = S4[idx];
                    S4[idx]  = make_float4(v.x, v.y, v.w, v.z);
                }
                __syncthreads();
            }
            {   // i=15: ctrl bit0, tgt bit15: odd elements swap across bit15
                for (int h = tid; h < 8192; h += THREADS) {
                    int i1 = h;                    // bit15 of z is 0
                    int i2 = h ^ 0x2000;           // partner block, bit15 = 1
                    float4 a = S4[i1];
                    float4 c = S4[i2];
                    float t;
                    t = a.y; a.y = c.y; c.y = t;
                    t = a.w; a.w = c.w; c.w = t;
                    S4[i1] = a;
                    S4[i2] = c;
                }
                __syncthreads();
            }
        }

        // ---- Accumulate Z-expval contributions: sum_z sign_w(z) * S[z]^2
        for (int jj = 0; jj < DIM / 4 / THREADS; ++jj) {
            int blkz = tid + THREADS * jj;
            int z0   = blkz << 2;
            float4 v = S4[blkz];
            float p0 = v.x * v.x, p1 = v.y * v.y, p2 = v.z * v.z, p3 = v.w * v.w;
            float ps = p0 + p1 + p2 + p3;
            // wires 0..13: sign bit (15-w) >= 2, constant within the block
#pragma unroll
            for (int w = 0; w < 14; ++w)
                acc[w] += ((z0 >> (15 - w)) & 1) ? -ps : ps;
            acc[14] += (p0 + p1) - (p2 + p3);      // wire 14 <-> bit 1
            acc[15] += (p0 + p2) - (p1 + p3);      // wire 15 <-> bit 0
        }
        __syncthreads();   // before next part overwrites S
    }

    // ================= Reduce 512 threads -> 16 expvals =================
#pragma unroll
    for (int w = 0; w < 16; ++w) {
        float v = acc[w];
        v += __shfl_xor(v, 16, 32);
        v += __shfl_xor(v, 8, 32);
        v += __shfl_xor(v, 4, 32);
        v += __shfl_xor(v, 2, 32);
        v += __shfl_xor(v, 1, 32);
        acc[w] = v;
    }
    if (lane == 0) {
#pragma unroll
        for (int w = 0; w < 16; ++w) partial[wave][w] = acc[w];
    }
    __syncthreads();
    if (tid < 16) {
        float s = 0.0f;
#pragma unroll
        for (int q = 0; q < 16; ++q) s += partial[q][tid];
        out[b * N_QUBITS + tid] = s;
    }
}

extern "C" void kernel_launch(void* const* d_in, const int* in_sizes, int n_in,
                              void* d_out, int out_size, void* d_ws, size_t ws_size,
                              hipStream_t stream) {
    const float* x      = (const float*)d_in[0];   // (512, 16) float32
    const float* params = (const float*)d_in[1];   // (32,) float32
    float* out          = (float*)d_out;           // (512, 16) float32
    (void)in_sizes; (void)n_in; (void)out_size; (void)d_ws; (void)ws_size;
    qsim16_kernel<<<BATCH, THREADS, 0, stream>>>(x, params, out);
}